// ContextMemoryManager_68771016343926
// MI455X (gfx1250) — compile-verified
//
#include <hip/hip_runtime.h>
#include <math.h>
#include <stdint.h>

typedef __attribute__((ext_vector_type(2))) float v2f;
typedef __attribute__((ext_vector_type(8))) float v8f;

#define D_DIM 4096
#define H_DIM 128
#define N_SEG 100
#define TOPK  10

__device__ __forceinline__ float gelu_exact(float x) {
    // erf-based (exact) GELU, matching jax.nn.gelu(approximate=False)
    return 0.5f * x * (1.0f + erff(x * 0.7071067811865476f));
}

// ---------------------------------------------------------------------------
// Kernel 1: per-segment importance MLP + segment half of relevance projection.
//   imp[n]  = sigmoid( gelu(seg@iw1+ib1) @ iw2 + ib2 )
//   impw[n] = imp[n] * (0.5 + 0.5 * 0.95^(N - pos - 1))
//   shv[n,h]= (seg @ rw1[D:])[h] + rb1[h]
// ---------------------------------------------------------------------------
__global__ void __launch_bounds__(128)
seg_kernel(const float* __restrict__ seg_emb,
           const int*   __restrict__ positions,
           const float* __restrict__ iw1, const float* __restrict__ ib1,
           const float* __restrict__ iw2, const float* __restrict__ ib2,
           const float* __restrict__ rw1, const float* __restrict__ rb1,
           float* __restrict__ shv, float* __restrict__ imp,
           float* __restrict__ impw) {
    const int n = blockIdx.x;
    const int h = threadIdx.x;                 // 0..127
    const float* seg  = seg_emb + (size_t)n * D_DIM;
    const float* rw1s = rw1 + (size_t)D_DIM * H_DIM;   // segment half of rw1

    float a1 = 0.f, a2 = 0.f;
#pragma unroll 4
    for (int d = 0; d < D_DIM; ++d) {
        float s = seg[d];                      // uniform across block -> scalar
        a1 = fmaf(s, iw1 [(size_t)d * H_DIM + h], a1);
        a2 = fmaf(s, rw1s[(size_t)d * H_DIM + h], a2);
    }

    float part = gelu_exact(a1 + ib1[h]) * iw2[h];
#pragma unroll
    for (int off = 16; off > 0; off >>= 1) part += __shfl_xor(part, off, 32);

    __shared__ float red[4];
    const int wave = h >> 5, lane = h & 31;
    if (lane == 0) red[wave] = part;
    __syncthreads();
    if (h == 0) {
        float s  = red[0] + red[1] + red[2] + red[3] + ib2[0];
        float im = 1.0f / (1.0f + expf(-s));
        float pf = powf(0.95f, (float)(N_SEG - positions[n] - 1));
        imp[n]  = im;
        impw[n] = im * (0.5f + 0.5f * pf);
    }
    shv[n * H_DIM + h] = a2 + rb1[h];
}

// ---------------------------------------------------------------------------
// Kernel 2: qh = query[B,D] @ rw1[:D]  ->  [B, 128], fp32 WMMA 16x16x4.
// 1 WG = 256 thr = 8 waves; WG owns 32 rows (2 M-tiles per wave), wave w owns
// columns [16w, 16w+16). Each B fragment feeds 2 WMMAs (halves rw1 L2 reads).
// A tile (32 x 32) staged in LDS via CDNA5 async DMA (global->LDS, ASYNCcnt).
// Pitch 36 floats: 16B-aligned rows for b128 async writes, and 36*m mod 64 is
// distinct for m=0..15 -> conflict-free column reads of the A fragments.
// ---------------------------------------------------------------------------
#define APITCH 36

__global__ void __launch_bounds__(256)
qh_gemm_kernel(const float* __restrict__ query,
               const float* __restrict__ rw1,
               float* __restrict__ qh) {
    const int m0   = blockIdx.x << 5;          // 32 rows per WG
    const int tid  = threadIdx.x;
    const int wave = tid >> 5;
    const int lane = tid & 31;
    const int n0   = wave << 4;

    __shared__ float At[32][APITCH];

    v8f c0 = {};
    v8f c1 = {};
    const int am    = lane & 15;               // A row (M) within tile
    const int akoff = (lane >> 4) << 1;        // A/B k sub-offset (0 or 2)
    const int bn    = n0 + (lane & 15);        // B column (N)

    // per-thread async staging coords: 32 rows x 32 k, 16B (4 floats) each
    const int r  = tid >> 3;                   // 0..31
    const int c4 = (tid & 7) << 2;             // 0,4,...,28
    const uint32_t ldsa =
        (uint32_t)(uintptr_t)(&At[r][c4]);     // low 32 bits == LDS byte offset
    const float* gA = query + (size_t)(m0 + r) * D_DIM + c4;

    for (int k0 = 0; k0 < D_DIM; k0 += 32) {
        __syncthreads();                       // previous tile fully consumed
        {   // CDNA5 async DMA: global -> LDS, tracked by ASYNCcnt
            uint64_t ga = (uint64_t)(uintptr_t)(gA + k0);
            asm volatile("global_load_async_to_lds_b128 %0, %1, off"
                         :: "v"(ldsa), "v"(ga) : "memory");
            asm volatile("s_wait_asynccnt 0x0" ::: "memory");
        }
        __syncthreads();                       // tile visible to all waves
#pragma unroll
        for (int kk = 0; kk < 32; kk += 4) {
            v2f a0, a1, b;
            // B frag: lane L holds B[(L/16)*2 + j][n]; loaded once, used twice
            const int bk = k0 + kk + akoff;
            b[0] = rw1[(size_t)bk       * H_DIM + bn];
            b[1] = rw1[(size_t)(bk + 1) * H_DIM + bn];
            // A frags: lane L holds A[m][(L/16)*2 + j] for j=0,1
            a0[0] = At[am][kk + akoff];
            a0[1] = At[am][kk + akoff + 1];
            a1[0] = At[16 + am][kk + akoff];
            a1[1] = At[16 + am][kk + akoff + 1];
            c0 = __builtin_amdgcn_wmma_f32_16x16x4_f32(
                     false, a0, false, b, (short)0, c0, false, false);
            c1 = __builtin_amdgcn_wmma_f32_16x16x4_f32(
                     false, a1, false, b, (short)0, c1, false, false);
        }
    }

    // C/D layout: VGPR i -> M = (lane>=16 ? 8 : 0) + i, N = lane%16
    const int sm = m0 + ((lane >> 4) << 3);
    const int sn = n0 + (lane & 15);
#pragma unroll
    for (int i = 0; i < 8; ++i)
        qh[(size_t)(sm + i) * H_DIM + sn] = c0[i];
#pragma unroll
    for (int i = 0; i < 8; ++i)
        qh[(size_t)(sm + 16 + i) * H_DIM + sn] = c1[i];
}

// ---------------------------------------------------------------------------
// Kernel 3: per batch row: relevance over N segments, top-10, weighted gather.
// ---------------------------------------------------------------------------
__global__ void __launch_bounds__(128)
score_topk_context_kernel(const float* __restrict__ query,
                          const float* __restrict__ seg_emb,
                          const float* __restrict__ qh,
                          const float* __restrict__ shv,
                          const float* __restrict__ imp,
                          const float* __restrict__ impw,
                          const float* __restrict__ rw2,
                          const float* __restrict__ rb2,
                          float* __restrict__ out) {
    const int b    = blockIdx.x;
    const int tid  = threadIdx.x;
    const int wave = tid >> 5;
    const int lane = tid & 31;

    __shared__ float s_w[N_SEG];       // imp * rel            (weights)
    __shared__ float s_score[N_SEG];   // imp * posf * rel     (ranking)
    __shared__ float s_rv[4];
    __shared__ int   s_ri[4];
    __shared__ int   s_sel[TOPK];
    __shared__ float s_wn[TOPK];

    float q[4], r2[4];
#pragma unroll
    for (int j = 0; j < 4; ++j) {
        const int hh = lane + 32 * j;
        q[j]  = qh[(size_t)b * H_DIM + hh];
        r2[j] = rw2[hh];
    }
    const float bias2 = rb2[0];

    // each wave handles every 4th segment; no cross-wave sync in the loop
    for (int n = wave; n < N_SEG; n += 4) {
        float acc = 0.f;
#pragma unroll
        for (int j = 0; j < 4; ++j) {
            const int hh = lane + 32 * j;
            acc += gelu_exact(q[j] + shv[n * H_DIM + hh]) * r2[j];
        }
#pragma unroll
        for (int off = 16; off > 0; off >>= 1) acc += __shfl_xor(acc, off, 32);
        if (lane == 0) {
            float rel  = 1.0f / (1.0f + expf(-(acc + bias2)));
            s_w[n]     = imp[n]  * rel;
            s_score[n] = impw[n] * rel;
        }
    }
    __syncthreads();

    // top-10 via 10 rounds of parallel argmax (mask winners with -inf)
    for (int k = 0; k < TOPK; ++k) {
        float v   = (tid < N_SEG) ? s_score[tid] : -3.4e38f;
        int   idx = tid;
#pragma unroll
        for (int off = 16; off > 0; off >>= 1) {
            float ov = __shfl_xor(v, off, 32);
            int   oi = __shfl_xor(idx, off, 32);
            if (ov > v) { v = ov; idx = oi; }
        }
        if (lane == 0) { s_rv[wave] = v; s_ri[wave] = idx; }
        __syncthreads();
        if (tid == 0) {
            float bv = s_rv[0]; int bi = s_ri[0];
            for (int wv = 1; wv < 4; ++wv)
                if (s_rv[wv] > bv) { bv = s_rv[wv]; bi = s_ri[wv]; }
            s_sel[k]    = bi;
            s_score[bi] = -3.4e38f;
        }
        __syncthreads();
    }

    if (tid == 0) {
        float s = 0.f;
        for (int k = 0; k < TOPK; ++k) s += s_w[s_sel[k]];
        const float inv = 1.0f / (s + 1e-8f);     // * BUDGET (=1.0)
        for (int k = 0; k < TOPK; ++k) s_wn[k] = s_w[s_sel[k]] * inv;
    }
    __syncthreads();

    int   sel[TOPK];
    float wn [TOPK];
#pragma unroll
    for (int k = 0; k < TOPK; ++k) { sel[k] = s_sel[k]; wn[k] = s_wn[k]; }

    for (int d = tid; d < D_DIM; d += 128) {
        float acc = query[(size_t)b * D_DIM + d];
#pragma unroll
        for (int k = 0; k < TOPK; ++k)
            acc = fmaf(wn[k], seg_emb[(size_t)sel[k] * D_DIM + d], acc);
        out[(size_t)b * D_DIM + d] = acc;
    }
}

// ---------------------------------------------------------------------------
extern "C" void kernel_launch(void* const* d_in, const int* in_sizes, int n_in,
                              void* d_out, int out_size, void* d_ws, size_t ws_size,
                              hipStream_t stream) {
    const float* query     = (const float*)d_in[0];
    const float* seg_emb   = (const float*)d_in[1];
    const int*   positions = (const int*)  d_in[2];
    const float* iw1       = (const float*)d_in[3];
    const float* ib1       = (const float*)d_in[4];
    const float* iw2       = (const float*)d_in[5];
    const float* ib2       = (const float*)d_in[6];
    const float* rw1       = (const float*)d_in[7];
    const float* rb1       = (const float*)d_in[8];
    const float* rw2       = (const float*)d_in[9];
    const float* rb2       = (const float*)d_in[10];
    float* out = (float*)d_out;

    const int B = in_sizes[0] / D_DIM;         // 4096

    // workspace layout (floats): qh[B*128] | shv[100*128] | imp[128] | impw[128]
    float* ws   = (float*)d_ws;
    float* qh   = ws;
    float* shv  = qh  + (size_t)B * H_DIM;
    float* imp  = shv + (size_t)N_SEG * H_DIM;
    float* impw = imp + 128;

    hipLaunchKernelGGL(seg_kernel, dim3(N_SEG), dim3(H_DIM), 0, stream,
                       seg_emb, positions, iw1, ib1, iw2, ib2, rw1, rb1,
                       shv, imp, impw);
    hipLaunchKernelGGL(qh_gemm_kernel, dim3(B / 32), dim3(256), 0, stream,
                       query, rw1, qh);
    hipLaunchKernelGGL(score_topk_context_kernel, dim3(B), dim3(128), 0, stream,
                       query, seg_emb, qh, shv, imp, impw, rw2, rb2, out);
}